// GINConv_57526791963072
// MI455X (gfx1250) — compile-verified
//
#include <hip/hip_runtime.h>
#include <hip/hip_bf16.h>

typedef __attribute__((ext_vector_type(2))) float v2f;
typedef __attribute__((ext_vector_type(8))) float v8f;

#define TB 256
#define BN_EPS 1e-5f

// ---------------------------------------------------------------- zero scratch
__global__ void k_zero(float* p, long n) {
    long i = (long)blockIdx.x * blockDim.x + threadIdx.x;
    if (i < n) p[i] = 0.0f;
}

// ------------------------------------------------- edge scatter (u_add_e, sum)
// one wave-half-pair per edge: 32 lanes x float2 = 64 floats
__global__ void k_edge(const float* __restrict__ node,
                       const float* __restrict__ edge,
                       const int* __restrict__ src,
                       const int* __restrict__ dst,
                       float* __restrict__ agg,
                       float* __restrict__ deg,
                       int E) {
    long tid = (long)blockIdx.x * blockDim.x + threadIdx.x;
    long e = tid >> 5;
    if (e >= E) return;
    int lane = (int)(tid & 31);
    int s = src[e], d = dst[e];
    float2 nv = ((const float2*)node)[(size_t)s * 32 + lane];
    float2 ev = ((const float2*)edge)[(size_t)e * 32 + lane];
    float2 m = make_float2(nv.x + ev.x, nv.y + ev.y);
    float* ap = &agg[(size_t)d * 64 + lane * 2];
    atomicAdd(ap, m.x);
    atomicAdd(ap + 1, m.y);
    if (lane == 0) atomicAdd(&deg[d], 1.0f);
}

// ---------------------------------- combine + GEMM1 (64->128) + column stats
// block: 256 thr / 8 waves; 128 rows per block; wave w -> rows w*16..w*16+15,
// all 8 column tiles (128 cols). K = 64 -> 16 wmma k-steps.
__global__ void k_gemm1(const float* __restrict__ node,
                        const float* __restrict__ agg,
                        const float* __restrict__ deg,
                        const float* __restrict__ epsp,
                        const float* __restrict__ W1,
                        const float* __restrict__ b1,
                        float* __restrict__ y1,
                        float* __restrict__ s1,
                        float* __restrict__ q1,
                        int N) {
    __shared__ float sh[128 * 65];           // h0 tile, padded stride 65
    const int tid = threadIdx.x;
    const int rb = blockIdx.x * 128;
    const float e1 = 1.0f + epsp[0];

    // stage h0 = (1+eps)*node + agg/max(deg,1)
    for (int idx = tid; idx < 128 * 64; idx += TB) {
        int r = idx >> 6, c = idx & 63;
        int g = rb + r;
        float v = 0.0f;
        if (g < N) {
            float inv = 1.0f / fmaxf(deg[g], 1.0f);
            v = e1 * node[(size_t)g * 64 + c] + agg[(size_t)g * 64 + c] * inv;
        }
        sh[r * 65 + c] = v;
    }
    __syncthreads();

    const int wave = tid >> 5, lane = tid & 31;
    const int m = lane & 15, kh = lane >> 4;
    const float* arow = &sh[(wave * 16 + m) * 65];

    v8f acc[8] = {};
    for (int kb = 0; kb < 64; kb += 4) {
        int k0 = kb + 2 * kh;
        v2f a;
        a.x = arow[k0];
        a.y = arow[k0 + 1];
#pragma unroll
        for (int t = 0; t < 8; ++t) {
            v2f b;
            b.x = W1[k0 * 128 + t * 16 + m];
            b.y = W1[(k0 + 1) * 128 + t * 16 + m];
            acc[t] = __builtin_amdgcn_wmma_f32_16x16x4_f32(
                false, a, false, b, (short)0, acc[t], false, false);
        }
    }

    // epilogue: +bias, store pre-BN activations, accumulate column sums/sumsq
#pragma unroll
    for (int t = 0; t < 8; ++t) {
        int col = t * 16 + m;
        float bias = b1[col];
        float s = 0.0f, q = 0.0f;
#pragma unroll
        for (int r = 0; r < 8; ++r) {
            int g = rb + wave * 16 + kh * 8 + r;
            float v = acc[t][r] + bias;
            if (g < N) {
                y1[(size_t)g * 128 + col] = v;
                s += v;
                q += v * v;
            }
        }
        atomicAdd(&s1[col], s);
        atomicAdd(&q1[col], q);
    }
}

// ------------------------------------------- BN finalize: scale/shift per col
__global__ void k_bnfin(const float* __restrict__ s,
                        const float* __restrict__ q,
                        const float* __restrict__ gamma,
                        const float* __restrict__ beta,
                        float* __restrict__ scale,
                        float* __restrict__ shift,
                        int C, float Nf) {
    int c = threadIdx.x;
    if (c >= C) return;
    float mean = s[c] / Nf;
    float var = q[c] / Nf - mean * mean;
    float sc = gamma[c] * rsqrtf(var + BN_EPS);
    scale[c] = sc;
    shift[c] = beta[c] - mean * sc;
}

// ----------------------- BN1+ReLU + GEMM2 (128->64) + column stats
// block: 256 thr / 8 waves; 64 rows per block; wave w -> rows (w&3)*16,
// col tiles (w>>2)*2 + {0,1}. K = 128 -> 32 wmma k-steps.
__global__ void k_gemm2(const float* __restrict__ y1,
                        const float* __restrict__ scale1,  // [128] then shift at +128
                        const float* __restrict__ W2,
                        const float* __restrict__ b2,
                        float* __restrict__ y2,
                        float* __restrict__ s2,
                        float* __restrict__ q2,
                        int N) {
    __shared__ float sh[64 * 129];           // relu(bn(y1)) tile, padded stride
    const int tid = threadIdx.x;
    const int rb = blockIdx.x * 64;

    for (int idx = tid; idx < 64 * 128; idx += TB) {
        int r = idx >> 7, c = idx & 127;
        int g = rb + r;
        float v = 0.0f;
        if (g < N)
            v = fmaxf(y1[(size_t)g * 128 + c] * scale1[c] + scale1[128 + c], 0.0f);
        sh[r * 129 + c] = v;
    }
    __syncthreads();

    const int wave = tid >> 5, lane = tid & 31;
    const int m = lane & 15, kh = lane >> 4;
    const int rloc = (wave & 3) * 16;
    const int cb = (wave >> 2) * 32;
    const float* arow = &sh[(rloc + m) * 129];

    v8f acc[2] = {};
    for (int kb = 0; kb < 128; kb += 4) {
        int k0 = kb + 2 * kh;
        v2f a;
        a.x = arow[k0];
        a.y = arow[k0 + 1];
#pragma unroll
        for (int t = 0; t < 2; ++t) {
            v2f b;
            b.x = W2[k0 * 64 + cb + t * 16 + m];
            b.y = W2[(k0 + 1) * 64 + cb + t * 16 + m];
            acc[t] = __builtin_amdgcn_wmma_f32_16x16x4_f32(
                false, a, false, b, (short)0, acc[t], false, false);
        }
    }

#pragma unroll
    for (int t = 0; t < 2; ++t) {
        int col = cb + t * 16 + m;
        float bias = b2[col];
        float s = 0.0f, q = 0.0f;
#pragma unroll
        for (int r = 0; r < 8; ++r) {
            int g = rb + rloc + kh * 8 + r;
            float v = acc[t][r] + bias;
            if (g < N) {
                y2[(size_t)g * 64 + col] = v;
                s += v;
                q += v * v;
            }
        }
        atomicAdd(&s2[col], s);
        atomicAdd(&q2[col], q);
    }
}

// --------------------------------------------- final BN2 + ReLU, elementwise
__global__ void k_out(const float* __restrict__ y2,
                      const float* __restrict__ scale2,  // [64] then shift at +64
                      float* __restrict__ out,
                      long total) {
    long i = (long)blockIdx.x * blockDim.x + threadIdx.x;
    if (i >= total) return;
    int c = (int)(i & 63);
    out[i] = fmaxf(y2[i] * scale2[c] + scale2[64 + c], 0.0f);
}

// -----------------------------------------------------------------------------
extern "C" void kernel_launch(void* const* d_in, const int* in_sizes, int n_in,
                              void* d_out, int out_size, void* d_ws, size_t ws_size,
                              hipStream_t stream) {
    const float* node = (const float*)d_in[0];
    const float* edge = (const float*)d_in[1];
    const int* src    = (const int*)d_in[2];
    const int* dst    = (const int*)d_in[3];
    const float* eps  = (const float*)d_in[4];
    const float* W1   = (const float*)d_in[5];
    const float* b1   = (const float*)d_in[6];
    const float* g1   = (const float*)d_in[7];
    const float* be1  = (const float*)d_in[8];
    const float* W2   = (const float*)d_in[9];
    const float* b2   = (const float*)d_in[10];
    const float* g2   = (const float*)d_in[11];
    const float* be2  = (const float*)d_in[12];
    float* out = (float*)d_out;

    const int N = in_sizes[0] / 64;
    const int E = in_sizes[2];

    // workspace layout (floats)
    float* ws = (float*)d_ws;
    size_t off = 0;
    float* agg = ws + off;  off += (size_t)N * 64;
    float* deg = ws + off;  off += (size_t)N;
    float* s1  = ws + off;  off += 128;
    float* q1  = ws + off;  off += 128;
    float* p1  = ws + off;  off += 256;   // scale1[128], shift1[128]
    float* s2  = ws + off;  off += 64;
    float* q2  = ws + off;  off += 64;
    float* p2  = ws + off;  off += 128;   // scale2[64], shift2[64]
    size_t zero_count = off;              // everything above must start at 0
    float* y1  = ws + off;  off += (size_t)N * 128;
    float* y2  = ws + off;  off += (size_t)N * 64;

    // 1) zero accumulation buffers
    {
        long n = (long)zero_count;
        int blocks = (int)((n + TB - 1) / TB);
        k_zero<<<blocks, TB, 0, stream>>>(ws, n);
    }
    // 2) edge scatter
    {
        long threads = (long)E * 32;
        int blocks = (int)((threads + TB - 1) / TB);
        k_edge<<<blocks, TB, 0, stream>>>(node, edge, src, dst, agg, deg, E);
    }
    // 3) combine + GEMM1 + stats
    {
        int blocks = (N + 127) / 128;
        k_gemm1<<<blocks, TB, 0, stream>>>(node, agg, deg, eps, W1, b1, y1, s1, q1, N);
    }
    // 4) BN1 finalize
    k_bnfin<<<1, 128, 0, stream>>>(s1, q1, g1, be1, p1, p1 + 128, 128, (float)N);
    // 5) BN1+ReLU + GEMM2 + stats
    {
        int blocks = (N + 63) / 64;
        k_gemm2<<<blocks, TB, 0, stream>>>(y1, p1, W2, b2, y2, s2, q2, N);
    }
    // 6) BN2 finalize
    k_bnfin<<<1, 64, 0, stream>>>(s2, q2, g2, be2, p2, p2 + 64, 64, (float)N);
    // 7) BN2 + ReLU -> out
    {
        long total = (long)N * 64;
        int blocks = (int)((total + TB - 1) / TB);
        k_out<<<blocks, TB, 0, stream>>>(y2, p2, out, total);
    }
}